// wo4_Transformer_Model_38783554683268
// MI455X (gfx1250) — compile-verified
//
#include <hip/hip_runtime.h>
#include <math.h>

typedef float v2f __attribute__((ext_vector_type(2)));
typedef float v8f __attribute__((ext_vector_type(8)));

#define Gn   8
#define Dn   16
#define Ln   1024
#define Bn   16
#define En   8
#define OUTn 64
#define KSn  3
#define LPn  1022                    // L - KS + 1
#define NTILES 16                    // ceil(1022/64) L-tiles of 64
#define CF_N (Bn*Gn*OUTn*LPn)        // combine_features element count
#define PRS  132                     // pair-row stride (64 cols * 2, +4 pad, even)

// branch-free tanh via hardware exp: tanh(x) = (e^2x - 1)/(e^2x + 1)
__device__ __forceinline__ float fast_tanh(float v) {
    float c = fminf(fmaxf(v, -15.f), 15.f);
    float e = __expf(2.f * c);
    return __fdividef(e - 1.f, e + 1.f);
}

// ---------------------------------------------------------------------------
// Kernel 1: gating (softmax -> top-4 -> normalized gates), loss, gates_out.
// One block of 128 threads; thread t handles (g = t/16, b = t%16).
// ---------------------------------------------------------------------------
__device__ __forceinline__ float cv_squared8(const float* v) {
    float mean = 0.f;
    for (int i = 0; i < 8; ++i) mean += v[i];
    mean *= 0.125f;
    float var = 0.f;
    for (int i = 0; i < 8; ++i) { float d = v[i] - mean; var += d * d; }
    var *= (1.0f / 7.0f);            // ddof=1
    return var / (mean * mean + 1e-10f);
}

__global__ __launch_bounds__(128) void gate_kernel(
    const float* __restrict__ x, const float* __restrict__ w_gate,
    float* __restrict__ gates_ws, float* __restrict__ out)
{
    __shared__ float sG[Gn][Bn][En];
    __shared__ float sImp[Gn][En];
    __shared__ float sLoad[Gn][En];
    __shared__ float sLoss[Gn];

    const int t = threadIdx.x;
    const int g = t >> 4, b = t & 15;

    float acc[En];
    for (int e = 0; e < En; ++e) acc[e] = 0.f;

    const float* xb  = x + ((size_t)(b * 128 + g * Dn)) * Ln;
    const float* wgg = w_gate + g * 80 * En;
    for (int c = 0; c < Dn; ++c) {
        for (int tt = 0; tt < 5; ++tt) {
            float gi = xb[c * Ln + (Ln - 6 + tt)];   // time steps L-6 .. L-2
            int j = c * 5 + tt;
            for (int e = 0; e < En; ++e) acc[e] += gi * wgg[j * En + e];
        }
    }
    // softmax
    float mx = acc[0];
    for (int e = 1; e < En; ++e) mx = fmaxf(mx, acc[e]);
    float p[En], s = 0.f;
    for (int e = 0; e < En; ++e) { p[e] = __expf(acc[e] - mx); s += p[e]; }
    float invs = 1.f / s;
    for (int e = 0; e < En; ++e) p[e] *= invs;

    // top-4 (stable: first occurrence wins on ties, matching lax.top_k)
    float v[En], gts[En];
    for (int e = 0; e < En; ++e) { v[e] = p[e]; gts[e] = 0.f; }
    int   sel[4]; float val[4]; float sum4 = 0.f;
    for (int it = 0; it < 4; ++it) {
        float best = -1.f; int bi = 0;
        for (int e = 0; e < En; ++e) if (v[e] > best) { best = v[e]; bi = e; }
        sel[it] = bi; val[it] = best; sum4 += best; v[bi] = -1.f;
    }
    float invn = 1.f / (sum4 + 1e-6f);
    for (int it = 0; it < 4; ++it) gts[sel[it]] = val[it] * invn;

    for (int e = 0; e < En; ++e) {
        sG[g][b][e] = gts[e];
        gates_ws[(g * Bn + b) * En + e] = gts[e];
        out[CF_N + 1 + (b * En + e) * Gn + g] = gts[e];   // gates_out (B,E,G)
    }
    __syncthreads();

    if (t < 64) {
        int gg = t >> 3, e = t & 7;
        float imp = 0.f, ld = 0.f;
        for (int bb = 0; bb < Bn; ++bb) {
            float gv = sG[gg][bb][e];
            imp += gv;
            if (gv > 0.f) ld += 1.f;
        }
        sImp[gg][e] = imp; sLoad[gg][e] = ld;
    }
    __syncthreads();
    if (t < Gn) sLoss[t] = cv_squared8(sImp[t]) + cv_squared8(sLoad[t]);
    __syncthreads();
    if (t == 0) {
        float tot = 0.f;
        for (int gg = 0; gg < Gn; ++gg) tot += sLoss[gg];
        out[CF_N] = 0.01f * tot;                         // out_loss
    }
}

// ---------------------------------------------------------------------------
// Kernel 2: fold gates into conv2 -> per-(g,b) eff_w[64][64], eff_b[64].
// One block per (g,b).
// ---------------------------------------------------------------------------
__global__ __launch_bounds__(256) void effw_kernel(
    const float* __restrict__ c2w, const float* __restrict__ c2b,
    const float* __restrict__ gates_ws,
    float* __restrict__ effw, float* __restrict__ effb)
{
    const int gb = blockIdx.x;
    const int g = gb >> 4;
    float gate[En];
    for (int e = 0; e < En; ++e) gate[e] = gates_ws[gb * En + e];

    const float* c2wg = c2w + (size_t)g * 512 * OUTn;    // (512, 64)
    float* ew = effw + (size_t)gb * (OUTn * OUTn);
    for (int i = threadIdx.x; i < OUTn * OUTn; i += 256) {
        int d = i >> 6, c = i & 63;
        float sum = 0.f;
        #pragma unroll
        for (int e = 0; e < En; ++e) sum += gate[e] * c2wg[(d * En + e) * OUTn + c];
        ew[i] = sum;
    }
    if (threadIdx.x < OUTn) {
        int d = threadIdx.x;
        float sum = 0.f;
        #pragma unroll
        for (int e = 0; e < En; ++e) sum += gate[e] * c2b[g * 512 + d * En + e];
        effb[gb * OUTn + d] = sum;
    }
}

// ---------------------------------------------------------------------------
// Kernel 3: fused conv1 (explicit im2col, K=48) + tanh + folded conv2 (K=64)
// via V_WMMA_F32_16X16X4_F32.  One block = one (g,b) x 64-wide L tile.
// B-matrices live in LDS in pair-interleaved layout [k/2][col][k%2] so each
// lane's B fragment is one contiguous 8B ds_load_b64 at a constant offset.
// 8 waves; wave w -> M-tile mt=w&3, N-tile pair (2*(w>>2), 2*(w>>2)+1).
// ---------------------------------------------------------------------------
__global__ __launch_bounds__(256) void moe_conv_kernel(
    const float* __restrict__ x,  const float* __restrict__ w1,
    const float* __restrict__ b1, const float* __restrict__ effw,
    const float* __restrict__ effb, float* __restrict__ out)
{
    __shared__ float sA1[OUTn * 48];   // conv1 weights as im2col A matrix (contiguous copy)
    __shared__ float sB1[24 * PRS];    // im2col x: 48 K-rows as 24 pair-rows x 64 cols x 2
    __shared__ float sW2[OUTn * 64];   // eff_w
    __shared__ float sH [32 * PRS];    // h tile: 64 rows as 32 pair-rows x 64 cols x 2

    const int t  = blockIdx.x;
    const int lt = t & 15;
    const int b  = (t >> 4) & 15;
    const int g  = t >> 8;
    const int l0 = lt * 64;
    const int tid = threadIdx.x;

    const float* w1g = w1 + (size_t)g * OUTn * Dn * KSn;          // (64,16,3) == A[64][48]
    for (int i = tid; i < OUTn * 48; i += 256) sA1[i] = w1g[i];
    const float* ewg = effw + (size_t)((g * Bn + b) * (OUTn * OUTn));
    for (int i = tid; i < OUTn * OUTn; i += 256) sW2[i] = ewg[i];

    // explicit im2col fill: B1[kappa][col] = x[c][l0+col+kk], kappa = c*3+kk
    const float* xb = x + (size_t)(b * 128 + g * Dn) * Ln;
    for (int i = tid; i < 24 * 128; i += 256) {
        int p = i >> 7, r = i & 127;        // pair-row, (col*2 + j)
        int col = r >> 1, j = r & 1;
        int kap = 2 * p + j;
        int c = kap / 3, kk = kap - 3 * c;
        int l = l0 + col + kk;
        sB1[p * PRS + r] = (l < Ln) ? xb[c * Ln + l] : 0.f;
    }
    __syncthreads();

    const int w    = tid >> 5;
    const int lane = tid & 31;
    const int hi   = lane >> 4, ln = lane & 15;
    const int mt   = w & 3;
    const int np   = w >> 2;
    const int col0 = (np * 2) * 16 + ln;      // N-tile 0 column
    const int col1 = (np * 2 + 1) * 16 + ln;  // N-tile 1 column

    v8f acc0 = {}, acc1 = {};

    // ---- Stage 1: h = conv1(x) over K = D*KS = 48 ----
    #pragma unroll
    for (int kb = 0; kb < 48; kb += 4) {
        v2f a  = *(const v2f*)&sA1[(mt * 16 + ln) * 48 + kb + 2 * hi];
        int pr = (kb >> 1) + hi;
        v2f f0 = *(const v2f*)&sB1[pr * PRS + col0 * 2];
        v2f f1 = *(const v2f*)&sB1[pr * PRS + col1 * 2];
        acc0 = __builtin_amdgcn_wmma_f32_16x16x4_f32(false, a, false, f0,
                                                     (short)0, acc0, false, false);
        acc1 = __builtin_amdgcn_wmma_f32_16x16x4_f32(false, a, false, f1,
                                                     (short)0, acc1, false, false);
    }

    // bias + tanh, write h tile to LDS (pair-interleaved)
    #pragma unroll
    for (int r = 0; r < 8; ++r) {
        int d = mt * 16 + 8 * hi + r;
        float bias = b1[g * OUTn + d];
        int rowbase = (d >> 1) * PRS + (d & 1);
        sH[rowbase + col0 * 2] = fast_tanh(acc0[r] + bias);
        sH[rowbase + col1 * 2] = fast_tanh(acc1[r] + bias);
    }
    __syncthreads();

    // ---- Stage 2: out = eff_w (64x64) x h, K = 64 ----
    #pragma unroll
    for (int r = 0; r < 8; ++r) { acc0[r] = 0.f; acc1[r] = 0.f; }
    #pragma unroll
    for (int kb = 0; kb < 64; kb += 4) {
        v2f a  = *(const v2f*)&sW2[(mt * 16 + ln) * 64 + kb + 2 * hi];
        int pr = (kb >> 1) + hi;
        v2f f0 = *(const v2f*)&sH[pr * PRS + col0 * 2];
        v2f f1 = *(const v2f*)&sH[pr * PRS + col1 * 2];
        acc0 = __builtin_amdgcn_wmma_f32_16x16x4_f32(false, a, false, f0,
                                                     (short)0, acc0, false, false);
        acc1 = __builtin_amdgcn_wmma_f32_16x16x4_f32(false, a, false, f1,
                                                     (short)0, acc1, false, false);
    }

    // epilogue: + eff_b, bounds-checked store into combine_features (B,G,OUT,Lp)
    const float* ebp = effb + (size_t)((g * Bn + b) * OUTn);
    const int lA = l0 + col0;
    const int lB = l0 + col1;
    #pragma unroll
    for (int r = 0; r < 8; ++r) {
        int d = mt * 16 + 8 * hi + r;
        float eb = ebp[d];
        size_t base = ((size_t)(b * Gn + g) * OUTn + d) * LPn;
        if (lA < LPn) out[base + lA] = acc0[r] + eb;
        if (lB < LPn) out[base + lB] = acc1[r] + eb;
    }
}

// ---------------------------------------------------------------------------
extern "C" void kernel_launch(void* const* d_in, const int* in_sizes, int n_in,
                              void* d_out, int out_size, void* d_ws, size_t ws_size,
                              hipStream_t stream) {
    const float* x      = (const float*)d_in[0];
    const float* w_gate = (const float*)d_in[1];
    const float* c1w    = (const float*)d_in[2];
    const float* c1b    = (const float*)d_in[3];
    const float* c2w    = (const float*)d_in[4];
    const float* c2b    = (const float*)d_in[5];
    float* out = (float*)d_out;
    float* ws  = (float*)d_ws;

    float* gates_ws = ws;                                  // 128*8      = 1024 f
    float* effw     = ws + 1024;                           // 128*64*64  = 524288 f
    float* effb     = ws + 1024 + 128 * OUTn * OUTn;       // 128*64     = 8192 f

    gate_kernel<<<1, 128, 0, stream>>>(x, w_gate, gates_ws, out);
    effw_kernel<<<Gn * Bn, 256, 0, stream>>>(c2w, c2b, gates_ws, effw, effb);
    moe_conv_kernel<<<Gn * Bn * NTILES, 256, 0, stream>>>(x, c1w, c1b, effw, effb, out);
}